// GATv2_79053168050940
// MI455X (gfx1250) — compile-verified
//
#include <hip/hip_runtime.h>
#include <math.h>

#define N_NODES 50000
#define E_RAW   400000
#define E_TOT   (E_RAW + N_NODES)
#define HEADS   4
#define MCHUNK  5              // M tiles per wave; 3125 = 625 * 5 exactly

typedef float v2f __attribute__((ext_vector_type(2)));
typedef float v8f __attribute__((ext_vector_type(8)));

// ---------------------------------------------------------------------------
// Y[M, Ncols] = X[M, 128] @ W[128, Ncols] + bias    (f32, exact)
// One wave sweeps MCHUNK 16-row M tiles of one 16-col slice. K-loop is
// software-pipelined one step ahead: A fragments (5x b64) and the shared B
// fragment (2x b32) for step kk+1 are issued before the 5 WMMAs of step kk,
// so v_wmma never sits behind an s_wait_loadcnt 0x0 drain.
// Fragment layouts per CDNA5 ISA 7.12.2:
//   A (16x4): lanes 0-15 -> M=lane, K={k,k+1}; lanes 16-31 -> M=lane-16, K={k+2,k+3}
//   B (4x16): mirrored with N across lanes
//   C/D: 8 VGPRs; lane<16 -> rows r0..r0+7, lane>=16 -> rows r0+8..r0+15
// ---------------------------------------------------------------------------
__global__ __launch_bounds__(256) void gemm_k128_wmma(
    const float* __restrict__ X, const float* __restrict__ W,
    const float* __restrict__ bias, float* __restrict__ Y,
    int M, int Ncols)
{
  const int lane   = threadIdx.x & 31;
  const int wave   = blockIdx.x * (blockDim.x >> 5) + (threadIdx.x >> 5);
  const int tilesN = Ncols >> 4;
  const int tileN  = wave % tilesN;
  const int chunk  = wave / tilesN;
  if (chunk * (16 * MCHUNK) >= M) return;        // wave-uniform: EXEC all-1 inside
  const int col0 = tileN << 4;
  const int half = lane >> 4, l16 = lane & 15;
  const int row_start = chunk * (16 * MCHUNK);

  const float* __restrict__ wcol = W + (size_t)(half * 2) * Ncols + col0 + l16;
  const float* __restrict__ xbase =
      X + (size_t)(row_start + l16) * 128 + half * 2;   // +mt*16*128 per tile

  v8f acc[MCHUNK];
#pragma unroll
  for (int mt = 0; mt < MCHUNK; ++mt) acc[mt] = (v8f){};

  // ---- prologue: fragments for kk = 0 ----
  v2f b_cur, a_cur[MCHUNK];
  b_cur.x = wcol[0];
  b_cur.y = wcol[(size_t)Ncols];
#pragma unroll
  for (int mt = 0; mt < MCHUNK; ++mt) {
    const float* xr = xbase + (size_t)mt * 16 * 128;
    a_cur[mt].x = xr[0];
    a_cur[mt].y = xr[1];
  }

#pragma unroll
  for (int kk = 0; kk < 32; ++kk) {
    v2f b_nxt, a_nxt[MCHUNK];
    if (kk < 31) {                 // issue next step's loads before the WMMAs
      const int kn = 4 * (kk + 1);
      b_nxt.x = wcol[(size_t)kn * Ncols];
      b_nxt.y = wcol[(size_t)(kn + 1) * Ncols];
#pragma unroll
      for (int mt = 0; mt < MCHUNK; ++mt) {
        const float* xr = xbase + (size_t)mt * 16 * 128;
        a_nxt[mt].x = xr[kn];
        a_nxt[mt].y = xr[kn + 1];
      }
    }
#pragma unroll
    for (int mt = 0; mt < MCHUNK; ++mt)
      acc[mt] = __builtin_amdgcn_wmma_f32_16x16x4_f32(
          false, a_cur[mt], false, b_cur, (short)0, acc[mt], false, false);
    if (kk < 31) {
      b_cur = b_nxt;
#pragma unroll
      for (int mt = 0; mt < MCHUNK; ++mt) a_cur[mt] = a_nxt[mt];
    }
  }

  const float bv = bias[col0 + l16];
#pragma unroll
  for (int mt = 0; mt < MCHUNK; ++mt) {
    const int row0 = row_start + mt * 16;
    float* __restrict__ out = Y + (size_t)(row0 + half * 8) * Ncols + col0 + l16;
#pragma unroll
    for (int r = 0; r < 8; ++r) out[(size_t)r * Ncols] = acc[mt][r] + bv;
  }
}

// ---- monotone float<->uint encoding so atomicMax(u32) == float max ---------
__device__ __forceinline__ unsigned fenc(float f) {
  unsigned u = __float_as_uint(f);
  return (u & 0x80000000u) ? ~u : (u | 0x80000000u);
}
__device__ __forceinline__ float fdec(unsigned u) {
  return (u & 0x80000000u) ? __uint_as_float(u ^ 0x80000000u)
                           : __uint_as_float(~u);
}

__device__ __forceinline__ void edge_src_dst(int e, const int* __restrict__ ei,
                                             int& src, int& dst) {
  if (e < E_RAW) { src = ei[e]; dst = ei[E_RAW + e]; }
  else           { src = dst = e - E_RAW; }
}

__device__ __forceinline__ float lrelu(float v) {
  return (v > 0.f) ? v : 0.2f * v;
}

// ---------------------------------------------------------------------------
// s[e,h] = att[h,:] . leakyrelu(xl[src,h,:] + xr[dst,h,:]);  atomic seg-max.
// One block per edge; 4 waves = 4 heads. C==128 path: one b128 load per lane
// from each of xl/xr/att (this phase is L2-gather bound; fewest/widest reqs).
// ---------------------------------------------------------------------------
__global__ __launch_bounds__(128) void edge_scores(
    const float* __restrict__ xl, const float* __restrict__ xr,
    const float* __restrict__ att, const int* __restrict__ ei,
    float* __restrict__ s, unsigned* __restrict__ menc, int C)
{
  const int e    = blockIdx.x;
  const int h    = threadIdx.x >> 5;
  const int lane = threadIdx.x & 31;
  int src, dst;
  edge_src_dst(e, ei, src, dst);
  const float* pl = xl + ((size_t)src * HEADS + h) * C;
  const float* pr = xr + ((size_t)dst * HEADS + h) * C;
  const float* pa = att + (size_t)h * C;
  float acc = 0.f;
  if (C == 128) {
    const float4 vl = ((const float4*)pl)[lane];
    const float4 vr = ((const float4*)pr)[lane];
    const float4 va = ((const float4*)pa)[lane];
    acc = lrelu(vl.x + vr.x) * va.x + lrelu(vl.y + vr.y) * va.y +
          lrelu(vl.z + vr.z) * va.z + lrelu(vl.w + vr.w) * va.w;
  } else {
#pragma unroll 2
    for (int c = lane; c < C; c += 32)
      acc += lrelu(pl[c] + pr[c]) * pa[c];
  }
#pragma unroll
  for (int off = 16; off > 0; off >>= 1) acc += __shfl_xor(acc, off, 32);
  if (lane == 0) {
    s[(size_t)e * HEADS + h] = acc;
    atomicMax(&menc[(size_t)dst * HEADS + h], fenc(acc));
  }
}

// p = exp(s - m[dst]) in place; z[dst,h] += p (atomic f32 add, L2 atomics)
__global__ __launch_bounds__(256) void edge_expsum(
    float* __restrict__ s, const unsigned* __restrict__ menc,
    float* __restrict__ z, const int* __restrict__ ei)
{
  const int t = blockIdx.x * blockDim.x + threadIdx.x;
  if (t >= E_TOT * HEADS) return;
  const int e = t >> 2, h = t & 3;
  int src, dst;
  edge_src_dst(e, ei, src, dst);
  (void)src;
  const float p = __expf(s[t] - fdec(menc[(size_t)dst * HEADS + h]));
  s[t] = p;
  atomicAdd(&z[(size_t)dst * HEADS + h], p);
}

// agg[dst,h,:] += (p/z[dst,h]) * xl[src,h,:]; one wave per (edge,head)
__global__ __launch_bounds__(128) void edge_aggregate(
    const float* __restrict__ p, const float* __restrict__ z,
    const float* __restrict__ xl, const int* __restrict__ ei,
    float* __restrict__ agg, int C)
{
  const int pair = blockIdx.x * 4 + (threadIdx.x >> 5);
  if (pair >= E_TOT * HEADS) return;       // wave-uniform
  const int lane = threadIdx.x & 31;
  const int e = pair >> 2, h = pair & 3;
  int src, dst;
  edge_src_dst(e, ei, src, dst);
  const float alpha = p[pair] / z[(size_t)dst * HEADS + h];
  const float* xs = xl + ((size_t)src * HEADS + h) * C;
  float* ad = agg + ((size_t)dst * HEADS + h) * C;
  for (int c = lane; c < C; c += 32)
    atomicAdd(&ad[c], alpha * xs[c]);
}

// out[n,c] = relu(mean_h(agg[n,h,c]) + bias[c])
__global__ __launch_bounds__(256) void finalize_relu(
    const float* __restrict__ agg, const float* __restrict__ bias,
    float* __restrict__ out, int C)
{
  const int t = blockIdx.x * blockDim.x + threadIdx.x;
  if (t >= N_NODES * C) return;
  const int n = t / C, c = t - n * C;
  const float* a = agg + (size_t)n * HEADS * C;
  float v = 0.25f * (a[c] + a[C + c] + a[2 * C + c] + a[3 * C + c]) + bias[c];
  out[t] = v > 0.f ? v : 0.f;
}

// Layer 2: t = mean_h + bias; out = log_softmax(t) over C=40. One wave/node.
__global__ __launch_bounds__(256) void finalize_logsoftmax(
    const float* __restrict__ agg, const float* __restrict__ bias,
    float* __restrict__ out)
{
  const int C = 40;
  const int n = blockIdx.x * 8 + (threadIdx.x >> 5);
  if (n >= N_NODES) return;                // wave-uniform
  const int lane = threadIdx.x & 31;
  const float* a = agg + (size_t)n * HEADS * C;
  const int c0 = lane, c1 = lane + 32;
  float t0 = 0.25f * (a[c0] + a[C + c0] + a[2 * C + c0] + a[3 * C + c0]) + bias[c0];
  float t1 = -INFINITY;
  if (c1 < C)
    t1 = 0.25f * (a[c1] + a[C + c1] + a[2 * C + c1] + a[3 * C + c1]) + bias[c1];
  float m = fmaxf(t0, t1);
#pragma unroll
  for (int off = 16; off > 0; off >>= 1) m = fmaxf(m, __shfl_xor(m, off, 32));
  float sum = __expf(t0 - m) + ((c1 < C) ? __expf(t1 - m) : 0.f);
#pragma unroll
  for (int off = 16; off > 0; off >>= 1) sum += __shfl_xor(sum, off, 32);
  const float l = m + __logf(sum);
  out[(size_t)n * C + c0] = t0 - l;
  if (c1 < C) out[(size_t)n * C + c1] = t1 - l;
}

// ---------------------------------------------------------------------------
extern "C" void kernel_launch(void* const* d_in, const int* in_sizes, int n_in,
                              void* d_out, int out_size, void* d_ws, size_t ws_size,
                              hipStream_t stream)
{
  const float* x  = (const float*)d_in[0];
  const int*   ei = (const int*)d_in[1];

  // workspace carve-out (floats)
  float* ws = (float*)d_ws;
  size_t o = 0;
  float*    xl   = ws + o; o += (size_t)N_NODES * 512;   // max Ncols = 512
  float*    xr   = ws + o; o += (size_t)N_NODES * 512;   // reused as agg
  float*    s    = ws + o; o += (size_t)E_TOT * HEADS;   // scores, then p
  unsigned* menc = (unsigned*)(ws + o); o += (size_t)N_NODES * HEADS;
  float*    z    = ws + o; o += (size_t)N_NODES * HEADS;
  float*    h0   = ws + o; o += (size_t)N_NODES * 128;
  float*    h1   = ws + o; o += (size_t)N_NODES * 128;
  (void)ws_size; (void)in_sizes; (void)n_in; (void)out_size;

  const float* xin = x;
  for (int layer = 0; layer < 3; ++layer) {
    const int C     = (layer == 2) ? 40 : 128;
    const int Ncols = HEADS * C;
    const float* Wl  = (const float*)d_in[2 + layer * 6 + 0];
    const float* bl  = (const float*)d_in[2 + layer * 6 + 1];
    const float* Wr  = (const float*)d_in[2 + layer * 6 + 2];
    const float* br  = (const float*)d_in[2 + layer * 6 + 3];
    const float* att = (const float*)d_in[2 + layer * 6 + 4];
    const float* bi  = (const float*)d_in[2 + layer * 6 + 5];

    const int tilesN = Ncols / 16;
    const int chunks = (N_NODES / 16) / MCHUNK;            // 625, exact
    const int waves  = tilesN * chunks;
    const int gGemm  = (waves + 7) / 8;                    // 8 waves / block
    gemm_k128_wmma<<<gGemm, 256, 0, stream>>>(xin, Wl, bl, xl, N_NODES, Ncols);
    gemm_k128_wmma<<<gGemm, 256, 0, stream>>>(xin, Wr, br, xr, N_NODES, Ncols);

    hipMemsetAsync(menc, 0, (size_t)N_NODES * HEADS * sizeof(unsigned), stream);
    hipMemsetAsync(z,    0, (size_t)N_NODES * HEADS * sizeof(float), stream);

    edge_scores<<<E_TOT, 128, 0, stream>>>(xl, xr, att, ei, s, menc, C);

    const int tEH = E_TOT * HEADS;
    edge_expsum<<<(tEH + 255) / 256, 256, 0, stream>>>(s, menc, z, ei);

    float* agg = xr;                        // xr dead after edge_scores
    hipMemsetAsync(agg, 0, (size_t)N_NODES * Ncols * sizeof(float), stream);
    edge_aggregate<<<(tEH + 3) / 4, 128, 0, stream>>>(s, z, xl, ei, agg, C);

    if (layer < 2) {
      float* hout = (layer == 0) ? h0 : h1;
      const int tNC = N_NODES * C;
      finalize_relu<<<(tNC + 255) / 256, 256, 0, stream>>>(agg, bi, hout, C);
      xin = hout;
    } else {
      finalize_logsoftmax<<<(N_NODES + 7) / 8, 256, 0, stream>>>(agg, bi, (float*)d_out);
    }
  }
}